// MultiHeadAttention_90675349553886
// MI455X (gfx1250) — compile-verified
//
#include <hip/hip_runtime.h>
#include <math.h>

typedef __attribute__((ext_vector_type(2))) float v2f;
typedef __attribute__((ext_vector_type(8))) float v8f;
typedef int gvec4i __attribute__((vector_size(16)));   // GCC-vector int4 (matches builtin)

#define WMMA_F32(a, b, c) \
    __builtin_amdgcn_wmma_f32_16x16x4_f32(false, (a), false, (b), (short)0, (c), false, false)

// ---- async global->LDS copies (CDNA5), guarded with sync fallback ---------
#if defined(__has_builtin)
#if __has_builtin(__builtin_amdgcn_global_load_async_to_lds_b32) && \
    __has_builtin(__builtin_amdgcn_global_load_async_to_lds_b128)
#define HAVE_ASYNC_LDS 1
#endif
#endif
#ifndef HAVE_ASYNC_LDS
#define HAVE_ASYNC_LDS 0
#endif

__device__ __forceinline__ void async_cp_b32(const float* g, float* l)
{
#if HAVE_ASYNC_LDS
    __builtin_amdgcn_global_load_async_to_lds_b32(
        (__attribute__((address_space(1))) int*)g,
        (__attribute__((address_space(3))) int*)l, 0, 0);
#else
    *l = *g;
#endif
}

__device__ __forceinline__ void async_cp_b128(const float* g, float* l)
{
#if HAVE_ASYNC_LDS
    __builtin_amdgcn_global_load_async_to_lds_b128(
        (__attribute__((address_space(1))) gvec4i*)g,
        (__attribute__((address_space(3))) gvec4i*)l, 0, 0);
#else
    *(float4*)l = *(const float4*)g;
#endif
}

__device__ __forceinline__ void wait_async0()
{
#if HAVE_ASYNC_LDS
#if __has_builtin(__builtin_amdgcn_s_wait_asynccnt)
    __builtin_amdgcn_s_wait_asynccnt(0);
#else
    asm volatile("s_wait_asynccnt 0" ::: "memory");
#endif
#endif
}

// ---------------------------------------------------------------------------
// GEMM: C[M,N] = A[M,K] @ B[K,N] + bias[N]   (fp32, WMMA 16x16x4)
// block tile 128(M) x 128(N), 8 waves (4Mx2N), wave tile 32x64 (2x4 frags).
// Double-buffered LDS slabs (K=16) staged with async global->LDS copies.
// ---------------------------------------------------------------------------
__global__ __launch_bounds__(256)
void gemm_bias_f32(const float* __restrict__ A, const float* __restrict__ B,
                   const float* __restrict__ bias, float* __restrict__ C,
                   int M, int N, int K)
{
    __shared__ float As[2 * 128 * 20];   // [buf][row][k] stride 20 (16B rows)
    __shared__ float Bs[2 * 128 * 20];   // [buf][n][k] transposed, stride 20

    const int tid  = threadIdx.x;
    const int lane = tid & 31;
    const int wave = tid >> 5;
    const int wm   = wave >> 1;        // 0..3
    const int wn   = wave & 1;         // 0..1
    const int hi   = (lane >> 4) & 1;
    const int l15  = lane & 15;

    const int base_m = blockIdx.y * 128;
    const int base_n = blockIdx.x * 128;

    v8f acc0[4] = {{}, {}, {}, {}};    // rows wm*32 + [0,16)
    v8f acc1[4] = {{}, {}, {}, {}};    // rows wm*32 + [16,32)

    // staging split: A via b128 (4 floats), B via b32 (transpose)
    const int a_q  = tid & 3;          // k-quad 0..3
    const int a_r0 = tid >> 2;         // 0..63
    const int b_n  = tid & 127;        // 0..127
    const int b_k0 = tid >> 7;         // 0..1

    auto stageAB = [&](int buf, int kt) {
        float* Ad = As + buf * 2560;
        float* Bd = Bs + buf * 2560;
        #pragma unroll
        for (int j = 0; j < 2; ++j) {
            const int r = a_r0 + j * 64;
            async_cp_b128(&A[(size_t)(base_m + r) * K + kt + a_q * 4],
                          &Ad[r * 20 + a_q * 4]);
        }
        #pragma unroll
        for (int j = 0; j < 8; ++j) {
            const int k = b_k0 + j * 2;
            async_cp_b32(&B[(size_t)(kt + k) * N + base_n + b_n],
                         &Bd[b_n * 20 + k]);
        }
    };

    stageAB(0, 0);
    wait_async0();
    __syncthreads();

    const int nslab = K >> 4;
    for (int s = 0; s < nslab; ++s) {
        const int buf = s & 1;
        if (s + 1 < nslab) stageAB(buf ^ 1, (s + 1) << 4);

        const float* Ab = As + buf * 2560;
        const float* Bb = Bs + buf * 2560;
        #pragma unroll
        for (int kk = 0; kk < 4; ++kk) {
            const int k0 = kk * 4 + hi * 2;
            v2f a0 = *(const v2f*)&Ab[(wm * 32 +      l15) * 20 + k0];
            v2f a1 = *(const v2f*)&Ab[(wm * 32 + 16 + l15) * 20 + k0];
            #pragma unroll
            for (int c = 0; c < 4; ++c) {
                v2f bc = *(const v2f*)&Bb[(wn * 64 + c * 16 + l15) * 20 + k0];
                acc0[c] = WMMA_F32(a0, bc, acc0[c]);
                acc1[c] = WMMA_F32(a1, bc, acc1[c]);
            }
        }
        wait_async0();
        __syncthreads();
    }

    // epilogue: C/D layout — VGPR v holds row (v + 8*hi), col = l15
    const int row0 = base_m + wm * 32 + hi * 8;
    #pragma unroll
    for (int c = 0; c < 4; ++c) {
        const int col = base_n + wn * 64 + c * 16 + l15;
        const float bv = bias[col];
        #pragma unroll
        for (int v = 0; v < 8; ++v) {
            C[(size_t)(row0 + v)      * N + col] = acc0[c][v] + bv;
            C[(size_t)(row0 + 16 + v) * N + col] = acc1[c][v] + bv;
        }
    }
}

// ---------------------------------------------------------------------------
// Flash attention (causal): qkv[B*T, 3072] -> Y[B*T, 1024]
// grid: (T/128, B*H), 8 waves/block, each wave owns 16 queries.
// ---------------------------------------------------------------------------
#define T_SEQ 2048
#define C3 3072

__global__ __launch_bounds__(256)
void attn_flash(const float* __restrict__ qkv, float* __restrict__ Y)
{
    __shared__ float Ks[16 * 68];   // [key][dim], stride 68 (16B rows for b128)
    __shared__ float Vs[64 * 18];   // transposed: [dim][key], stride 18
    __shared__ float Ps[8 * 288];   // per-wave P scratch, 16 rows x stride 18

    const int tid  = threadIdx.x;
    const int lane = tid & 31;
    const int wave = tid >> 5;
    const int hi   = (lane >> 4) & 1;
    const int l15  = lane & 15;

    const int b  = blockIdx.y >> 4;
    const int h  = blockIdx.y & 15;
    const int q0 = blockIdx.x * 128 + wave * 16;

    // preload Q fragments: 16 K-steps of 4, kept in registers (32 VGPRs)
    const float* Qrow = qkv + (size_t)(b * T_SEQ + q0 + l15) * C3 + h * 64;
    v2f aQ[16];
    #pragma unroll
    for (int s = 0; s < 16; ++s)
        aQ[s] = *(const v2f*)(Qrow + s * 4 + hi * 2);

    v8f o0 = {}, o1 = {}, o2 = {}, o3 = {};
    float m[8], l[8];
    #pragma unroll
    for (int v = 0; v < 8; ++v) { m[v] = -1e30f; l[v] = 0.0f; }

    // staging split: K via b128 (key = tid>>4, quad = tid&15),
    //                V via b32 transpose (dim = tid&63, key base = tid>>6)
    const int k_key = tid >> 4;
    const int k_q   = tid & 15;
    const int st_d  = tid & 63;
    const int st_k0 = tid >> 6;
    const int nkb   = blockIdx.x * 8 + 8;   // causal block count for this tile
    float* Pw = Ps + wave * 288;

    for (int kb = 0; kb < nkb; ++kb) {
        {
            const float* krow =
                qkv + (size_t)(b * T_SEQ + kb * 16 + k_key) * C3 + h * 64;
            async_cp_b128(krow + 1024 + k_q * 4, &Ks[k_key * 68 + k_q * 4]);
            #pragma unroll
            for (int j = 0; j < 4; ++j) {
                const int key = st_k0 + j * 4;
                const float* vrow =
                    qkv + (size_t)(b * T_SEQ + kb * 16 + key) * C3 + h * 64;
                async_cp_b32(vrow + 2048 + st_d, &Vs[st_d * 18 + key]);
            }
        }
        wait_async0();
        __syncthreads();

        if (kb * 16 <= q0) {   // wave-uniform causal block skip
            // S = Q @ K^T : 16 WMMAs over dk=64
            v8f s = {};
            #pragma unroll
            for (int kk = 0; kk < 16; ++kk) {
                const int k0 = kk * 4 + hi * 2;
                v2f bK = *(const v2f*)&Ks[l15 * 68 + k0];
                s = WMMA_F32(aQ[kk], bK, s);
            }

            // scale + causal mask + online softmax (C/D row = v + 8*hi)
            const int kcol = kb * 16 + l15;
            #pragma unroll
            for (int v = 0; v < 8; ++v) {
                float sv = s[v] * 0.125f;                 // 1/sqrt(64)
                const int qr = q0 + hi * 8 + v;
                if (kcol > qr) sv = -1e30f;
                float x = sv;                             // row max over 16 lanes
                x = fmaxf(x, __shfl_xor(x, 1, 32));
                x = fmaxf(x, __shfl_xor(x, 2, 32));
                x = fmaxf(x, __shfl_xor(x, 4, 32));
                x = fmaxf(x, __shfl_xor(x, 8, 32));
                const float mn = fmaxf(m[v], x);
                const float p  = __expf(sv - mn);
                float rs = p;                             // row sum over 16 lanes
                rs += __shfl_xor(rs, 1, 32);
                rs += __shfl_xor(rs, 2, 32);
                rs += __shfl_xor(rs, 4, 32);
                rs += __shfl_xor(rs, 8, 32);
                const float al = __expf(m[v] - mn);
                l[v] = l[v] * al + rs;
                m[v] = mn;
                o0[v] *= al; o1[v] *= al; o2[v] *= al; o3[v] *= al;
                Pw[(hi * 8 + v) * 18 + l15] = p;
            }

            // reload P as A-fragments (C/D layout -> A layout via LDS)
            v2f aP[4];
            #pragma unroll
            for (int kk = 0; kk < 4; ++kk)
                aP[kk] = *(const v2f*)&Pw[l15 * 18 + kk * 4 + hi * 2];

            // O += P @ V : 4 col tiles x 4 K-steps
            #pragma unroll
            for (int kk = 0; kk < 4; ++kk) {
                const int k0 = kk * 4 + hi * 2;
                v2f bv0 = *(const v2f*)&Vs[(     l15) * 18 + k0];
                v2f bv1 = *(const v2f*)&Vs[(16 + l15) * 18 + k0];
                v2f bv2 = *(const v2f*)&Vs[(32 + l15) * 18 + k0];
                v2f bv3 = *(const v2f*)&Vs[(48 + l15) * 18 + k0];
                o0 = WMMA_F32(aP[kk], bv0, o0);
                o1 = WMMA_F32(aP[kk], bv1, o1);
                o2 = WMMA_F32(aP[kk], bv2, o2);
                o3 = WMMA_F32(aP[kk], bv3, o3);
            }
        }
        __syncthreads();
    }

    // normalize and store: Y[b*T + q0 + r][h*64 + d]
    float* Yrow = Y + (size_t)(b * T_SEQ + q0) * 1024 + h * 64;
    #pragma unroll
    for (int v = 0; v < 8; ++v) {
        const float inv = 1.0f / l[v];
        const size_t off = (size_t)(hi * 8 + v) * 1024;
        Yrow[off +      l15] = o0[v] * inv;
        Yrow[off + 16 + l15] = o1[v] * inv;
        Yrow[off + 32 + l15] = o2[v] * inv;
        Yrow[off + 48 + l15] = o3[v] * inv;
    }
}

// ---------------------------------------------------------------------------
extern "C" void kernel_launch(void* const* d_in, const int* in_sizes, int n_in,
                              void* d_out, int out_size, void* d_ws, size_t ws_size,
                              hipStream_t stream)
{
    const float* x     = (const float*)d_in[0];   // [4,2048,1024]
    const float* Wqkv  = (const float*)d_in[1];   // [1024,3072]
    const float* bqkv  = (const float*)d_in[2];   // [3072]
    const float* Wproj = (const float*)d_in[3];   // [1024,1024]
    const float* bproj = (const float*)d_in[4];   // [1024]
    float* out = (float*)d_out;                   // [4,2048,1024]

    float* qkv = (float*)d_ws;                    // 8192 x 3072 fp32 (96 MB)
    float* y   = qkv + (size_t)8192 * 3072;       // 8192 x 1024 fp32 (32 MB)

    // 1) qkv = x @ W_qkv + b_qkv
    gemm_bias_f32<<<dim3(3072 / 128, 8192 / 128), 256, 0, stream>>>(
        x, Wqkv, bqkv, qkv, 8192, 3072, 1024);

    // 2) causal flash attention per (batch, head)
    attn_flash<<<dim3(T_SEQ / 128, 4 * 16), 256, 0, stream>>>(qkv, y);

    // 3) out = y @ W_proj + b_proj
    gemm_bias_f32<<<dim3(1024 / 128, 8192 / 128), 256, 0, stream>>>(
        y, Wproj, bproj, out, 8192, 1024, 1024);
}